// GAT_graph_branch_71012989272554
// MI455X (gfx1250) — compile-verified
//
#include <hip/hip_runtime.h>
#include <hip/hip_bf16.h>

// ---------------- problem constants ----------------
#define N_NODES 50000
#define N_EDGES 800000
#define N_ESL   (N_EDGES + N_NODES)   // edges + self loops
#define F_DIM   128                   // F_IN == HEADS*C_HID == 128
#define NF      ((size_t)N_NODES * F_DIM)
#define N_GRAPHS 32
#define NEG_SLOPE 0.2f

typedef float v2f __attribute__((ext_vector_type(2)));
typedef float v8f __attribute__((ext_vector_type(8)));

// ---------------- helpers ----------------
__device__ __forceinline__ unsigned int enc_f32(float f) {
    unsigned int u = __float_as_uint(f);
    return (u & 0x80000000u) ? ~u : (u | 0x80000000u);
}
__device__ __forceinline__ float dec_f32(unsigned int e) {
    return (e & 0x80000000u) ? __uint_as_float(e & 0x7FFFFFFFu) : __uint_as_float(~e);
}
__device__ __forceinline__ float lrelu(float v) {
    return v > 0.0f ? v : NEG_SLOPE * v;
}

// ---------------- dual GEMM via fp32 WMMA (16x16x4) ----------------
// Out{L,R}[N x 128] = X[N x 128] @ W{l,r}[128 x 128]
// grid.x = N/16 row tiles; block = 256 threads = 8 waves; wave w owns col tile w.
__global__ __launch_bounds__(256)
void gemm_dual_wmma(const float* __restrict__ X,
                    const float* __restrict__ Wl, const float* __restrict__ Wr,
                    float* __restrict__ OutL, float* __restrict__ OutR) {
    const int K = F_DIM, NC = F_DIM;
    const int m0   = blockIdx.x * 16;
    const int wave = threadIdx.x >> 5;
    const int n0   = wave * 16;
    const int lane = threadIdx.x & 31;
    const int half = lane >> 4;      // 0: K+0/K+1, 1: K+2/K+3 (A); cols via l
    const int l    = lane & 15;

    v8f accL = {};
    v8f accR = {};
    const float* xrow = X + (size_t)(m0 + l) * K;

    for (int k0 = 0; k0 < K; k0 += 4) {
        const int ka = k0 + half * 2;
        // A fragment: lane l holds row (m0+l), K = ka, ka+1
        v2f a = *(const v2f*)(xrow + ka);
        // B fragments: lane l holds col (n0+l), K = ka, ka+1
        v2f bl, br;
        bl.x = Wl[(size_t)ka * NC + n0 + l];
        bl.y = Wl[(size_t)(ka + 1) * NC + n0 + l];
        br.x = Wr[(size_t)ka * NC + n0 + l];
        br.y = Wr[(size_t)(ka + 1) * NC + n0 + l];
        accL = __builtin_amdgcn_wmma_f32_16x16x4_f32(false, a, false, bl,
                                                     (short)0, accL, false, false);
        accR = __builtin_amdgcn_wmma_f32_16x16x4_f32(false, a, false, br,
                                                     (short)0, accR, false, false);
    }
    // C/D layout: VGPR i -> row m0 + half*8 + i, col n0 + l
#pragma unroll
    for (int i = 0; i < 8; ++i) {
        const size_t row = (size_t)(m0 + half * 8 + i);
        OutL[row * NC + n0 + l] = accL[i];
        OutR[row * NC + n0 + l] = accR[i];
    }
}

// ---------------- GATv2 edge phase ----------------
// Pass 1: logits[e][h] = sum_c att[h][c] * leaky_relu(xr[dst] + xl[src]); segment-max into mEnc.
__global__ __launch_bounds__(256)
void edge_logits(const float* __restrict__ xl, const float* __restrict__ xr,
                 const int* __restrict__ ei, const float* __restrict__ att,
                 float* __restrict__ logitP, unsigned int* __restrict__ mEnc) {
    const int e = blockIdx.x * 8 + (threadIdx.x >> 5);
    if (e >= N_ESL) return;
    const int lane = threadIdx.x & 31;
    int s, d;
    if (e < N_EDGES) { s = ei[e]; d = ei[e + N_EDGES]; }
    else             { s = d = e - N_EDGES; }

    const float4 xlv = ((const float4*)(xl + (size_t)s * F_DIM))[lane];
    const float4 xrv = ((const float4*)(xr + (size_t)d * F_DIM))[lane];
    const float4 av  = ((const float4*)att)[lane];   // lanes 0-15: head0, 16-31: head1

    float sum = lrelu(xlv.x + xrv.x) * av.x + lrelu(xlv.y + xrv.y) * av.y +
                lrelu(xlv.z + xrv.z) * av.z + lrelu(xlv.w + xrv.w) * av.w;
    // reduce within each 16-lane half (wave32: one head per half-wave)
    sum += __shfl_xor(sum, 1, 32);
    sum += __shfl_xor(sum, 2, 32);
    sum += __shfl_xor(sum, 4, 32);
    sum += __shfl_xor(sum, 8, 32);
    if ((lane & 15) == 0) {
        const int h = lane >> 4;
        logitP[(size_t)e * 2 + h] = sum;
        atomicMax(&mEnc[(size_t)d * 2 + h], enc_f32(sum));
    }
}

// Pass 2: p = exp(logit - m[dst]); segment-sum into ssum; overwrite logits with p.
__global__ __launch_bounds__(256)
void edge_exp(float* __restrict__ logitP, const unsigned int* __restrict__ mEnc,
              const int* __restrict__ ei, float* __restrict__ ssum) {
    const int idx = blockIdx.x * blockDim.x + threadIdx.x;
    if (idx >= N_ESL * 2) return;
    const int e = idx >> 1, h = idx & 1;
    const int d = (e < N_EDGES) ? ei[e + N_EDGES] : e - N_EDGES;
    const float m = dec_f32(mEnc[(size_t)d * 2 + h]);
    const float p = __expf(logitP[idx] - m);
    logitP[idx] = p;
    atomicAdd(&ssum[(size_t)d * 2 + h], p);
}

// Pass 3: agg[dst] += xl[src] * (p / (ssum[dst] + 1e-16))
__global__ __launch_bounds__(256)
void edge_aggregate(const float* __restrict__ xl, const float* __restrict__ logitP,
                    const float* __restrict__ ssum, const int* __restrict__ ei,
                    float* __restrict__ agg) {
    const int e = blockIdx.x * 8 + (threadIdx.x >> 5);
    if (e >= N_ESL) return;
    const int lane = threadIdx.x & 31;
    int s, d;
    if (e < N_EDGES) { s = ei[e]; d = ei[e + N_EDGES]; }
    else             { s = d = e - N_EDGES; }
    const int h = lane >> 4;
    const float alpha = logitP[(size_t)e * 2 + h] /
                        (ssum[(size_t)d * 2 + h] + 1e-16f);
    const float4 v = ((const float4*)(xl + (size_t)s * F_DIM))[lane];
    float* o = agg + (size_t)d * F_DIM + lane * 4;
    atomicAdd(o + 0, v.x * alpha);
    atomicAdd(o + 1, v.y * alpha);
    atomicAdd(o + 2, v.z * alpha);
    atomicAdd(o + 3, v.w * alpha);
}

// y = elu(agg + bias)
__global__ __launch_bounds__(256)
void bias_elu(const float* __restrict__ agg, const float* __restrict__ b,
              float* __restrict__ y) {
    const size_t i = (size_t)blockIdx.x * blockDim.x + threadIdx.x;
    if (i >= NF) return;
    const float v = agg[i] + b[i & (F_DIM - 1)];
    y[i] = v > 0.0f ? v : (__expf(v) - 1.0f);
}

// ---------------- label propagation ----------------
__global__ __launch_bounds__(256)
void degree_k(const int* __restrict__ ei, float* __restrict__ deg) {
    const int e = blockIdx.x * blockDim.x + threadIdx.x;
    if (e >= N_EDGES) return;
    atomicAdd(&deg[ei[e + N_EDGES]], 1.0f);
}
__global__ __launch_bounds__(256)
void rsqrt_deg(const float* __restrict__ deg, float* __restrict__ dinv) {
    const int n = blockIdx.x * blockDim.x + threadIdx.x;
    if (n >= N_NODES) return;
    const float d = deg[n];
    dinv[n] = d > 0.0f ? rsqrtf(fmaxf(d, 1.0f)) : 0.0f;
}
__global__ __launch_bounds__(256)
void lp_scatter(const float* __restrict__ cur, const float* __restrict__ dinv,
                const int* __restrict__ ei, float* __restrict__ tmp) {
    const int e = blockIdx.x * 8 + (threadIdx.x >> 5);
    if (e >= N_EDGES) return;
    const int lane = threadIdx.x & 31;
    const int s = ei[e], d = ei[e + N_EDGES];
    const float w = dinv[s] * dinv[d];
    const float4 v = ((const float4*)(cur + (size_t)s * F_DIM))[lane];
    float* o = tmp + (size_t)d * F_DIM + lane * 4;
    atomicAdd(o + 0, v.x * w);
    atomicAdd(o + 1, v.y * w);
    atomicAdd(o + 2, v.z * w);
    atomicAdd(o + 3, v.w * w);
}
// out = clip(0.5*tmp + 0.5*y0, 0, 1)
__global__ __launch_bounds__(256)
void lp_update(const float* __restrict__ tmp, const float* __restrict__ y0,
               float* __restrict__ out) {
    const size_t i = (size_t)blockIdx.x * blockDim.x + threadIdx.x;
    if (i >= NF) return;
    const float v = 0.5f * tmp[i] + 0.5f * y0[i];
    out[i] = fminf(fmaxf(v, 0.0f), 1.0f);
}

// ---------------- pooling + MLP ----------------
__global__ __launch_bounds__(256)
void pool_count(const int* __restrict__ batch, float* __restrict__ cnt) {
    const int n = blockIdx.x * blockDim.x + threadIdx.x;
    if (n >= N_NODES) return;
    atomicAdd(&cnt[batch[n]], 1.0f);
}
__global__ __launch_bounds__(256)
void pool_accum(const float* __restrict__ x, const float* __restrict__ h1,
                const float* __restrict__ h2, const int* __restrict__ batch,
                float* __restrict__ sums) {
    const size_t idx = (size_t)blockIdx.x * blockDim.x + threadIdx.x;  // N * 96
    if (idx >= (size_t)N_NODES * 96) return;
    const int n = (int)(idx / 96);
    const int f = (int)(idx % 96) * 4;    // feature in [0,384)
    const float* srcp = (f < 128) ? (x  + (size_t)n * F_DIM + f)
                       : (f < 256) ? (h1 + (size_t)n * F_DIM + (f - 128))
                                   : (h2 + (size_t)n * F_DIM + (f - 256));
    const float4 v = *(const float4*)srcp;
    float* o = sums + (size_t)batch[n] * 384 + f;
    atomicAdd(o + 0, v.x);
    atomicAdd(o + 1, v.y);
    atomicAdd(o + 2, v.z);
    atomicAdd(o + 3, v.w);
}
__global__ __launch_bounds__(256)
void finalize_g(float* __restrict__ sums, const float* __restrict__ cnt) {
    const int i = blockIdx.x * blockDim.x + threadIdx.x;  // 32*384
    if (i >= N_GRAPHS * 384) return;
    sums[i] = sums[i] / fmaxf(cnt[i / 384], 1.0f);
}
__global__ __launch_bounds__(256)
void mlp1(const float* __restrict__ g, const float* __restrict__ W1,
          const float* __restrict__ b1, float* __restrict__ hid) {
    const int i = blockIdx.x * blockDim.x + threadIdx.x;  // 32*256
    if (i >= N_GRAPHS * 256) return;
    const int r = i >> 8, c = i & 255;
    float acc = b1[c];
    for (int k = 0; k < 384; ++k) acc += g[r * 384 + k] * W1[k * 256 + c];
    hid[i] = fmaxf(acc, 0.0f);
}
__global__ __launch_bounds__(256)
void mlp2(const float* __restrict__ hid, const float* __restrict__ W2,
          const float* __restrict__ b2, float* __restrict__ out) {
    const int i = blockIdx.x * blockDim.x + threadIdx.x;  // 32*128
    if (i >= N_GRAPHS * 128) return;
    const int r = i >> 7, c = i & 127;
    float acc = b2[c];
    for (int k = 0; k < 256; ++k) acc += hid[r * 256 + k] * W2[k * 128 + c];
    out[i] = acc;
}

// ---------------- launcher ----------------
extern "C" void kernel_launch(void* const* d_in, const int* in_sizes, int n_in,
                              void* d_out, int out_size, void* d_ws, size_t ws_size,
                              hipStream_t stream) {
    const float* x     = (const float*)d_in[0];
    const int*   ei    = (const int*)d_in[1];
    const int*   batch = (const int*)d_in[2];
    const float* Wl1   = (const float*)d_in[3];
    const float* Wr1   = (const float*)d_in[4];
    const float* att1  = (const float*)d_in[5];
    const float* b1    = (const float*)d_in[6];
    const float* Wl2   = (const float*)d_in[7];
    const float* Wr2   = (const float*)d_in[8];
    const float* att2  = (const float*)d_in[9];
    const float* b2    = (const float*)d_in[10];
    const float* mW1   = (const float*)d_in[11];
    const float* mb1   = (const float*)d_in[12];
    const float* mW2   = (const float*)d_in[13];
    const float* mb2   = (const float*)d_in[14];
    float* out = (float*)d_out;

    // workspace layout (floats)
    float* ws = (float*)d_ws;
    float* A    = ws;                 // xl          [N,128]
    float* B    = ws + NF;            // xr          [N,128]
    float* C    = ws + 2 * NF;        // agg / lp tmp[N,128]
    float* D    = ws + 3 * NF;        // h1 (post-LP)[N,128]
    float* Ebuf = ws + 4 * NF;        // h2 (post-LP)[N,128]
    float* Y    = ws + 5 * NF;        // elu output  [N,128]
    float*        P    = ws + 6 * NF;             // logits/p  [ESL,2]
    unsigned int* MENC = (unsigned int*)(P + (size_t)N_ESL * 2);  // [N,2]
    float* SSUM  = (float*)(MENC + (size_t)N_NODES * 2);          // [N,2]
    float* DEG   = SSUM + (size_t)N_NODES * 2;                    // [N]
    float* DINV  = DEG + N_NODES;                                 // [N]
    float* POOLS = DINV + N_NODES;                                // [32,384]
    float* CNT   = POOLS + N_GRAPHS * 384;                        // [32]
    float* HID   = CNT + N_GRAPHS;                                // [32,256]

    const dim3 blk(256);
    const int gGemm  = N_NODES / 16;                    // 3125
    const int gEdge  = (N_ESL + 7) / 8;                 // 8 edges / block
    const int gEdgeE = (N_EDGES + 7) / 8;
    const int gEH    = (N_ESL * 2 + 255) / 256;
    const int gNF    = (int)((NF + 255) / 256);
    const int gN     = (N_NODES + 255) / 256;
    const int gE     = (N_EDGES + 255) / 256;
    const int gPool  = (int)(((size_t)N_NODES * 96 + 255) / 256);

    // ---- zero accumulators (graph-capture-safe async memsets) ----
    hipMemsetAsync(C,     0, NF * sizeof(float), stream);
    hipMemsetAsync(MENC,  0, (size_t)N_NODES * 2 * sizeof(unsigned int), stream);
    hipMemsetAsync(SSUM,  0, (size_t)N_NODES * 2 * sizeof(float), stream);
    hipMemsetAsync(DEG,   0, (size_t)N_NODES * sizeof(float), stream);
    hipMemsetAsync(POOLS, 0, (size_t)N_GRAPHS * 384 * sizeof(float), stream);
    hipMemsetAsync(CNT,   0, (size_t)N_GRAPHS * sizeof(float), stream);

    // ---- GAT layer 1 ----
    gemm_dual_wmma<<<gGemm, blk, 0, stream>>>(x, Wl1, Wr1, A, B);
    edge_logits   <<<gEdge, blk, 0, stream>>>(A, B, ei, att1, P, MENC);
    edge_exp      <<<gEH,   blk, 0, stream>>>(P, MENC, ei, SSUM);
    edge_aggregate<<<gEdge, blk, 0, stream>>>(A, P, SSUM, ei, C);
    bias_elu      <<<gNF,   blk, 0, stream>>>(C, b1, Y);

    // ---- degree / norm (shared by both label-prop calls) ----
    degree_k <<<gE, blk, 0, stream>>>(ei, DEG);
    rsqrt_deg<<<gN, blk, 0, stream>>>(DEG, DINV);

    // ---- label prop 1 (2 steps, res = 0.5*Y) ----
    hipMemsetAsync(C, 0, NF * sizeof(float), stream);
    lp_scatter<<<gEdgeE, blk, 0, stream>>>(Y, DINV, ei, C);
    lp_update <<<gNF,    blk, 0, stream>>>(C, Y, D);
    hipMemsetAsync(C, 0, NF * sizeof(float), stream);
    lp_scatter<<<gEdgeE, blk, 0, stream>>>(D, DINV, ei, C);
    lp_update <<<gNF,    blk, 0, stream>>>(C, Y, D);

    // ---- GAT layer 2 ----
    hipMemsetAsync(C,    0, NF * sizeof(float), stream);
    hipMemsetAsync(MENC, 0, (size_t)N_NODES * 2 * sizeof(unsigned int), stream);
    hipMemsetAsync(SSUM, 0, (size_t)N_NODES * 2 * sizeof(float), stream);
    gemm_dual_wmma<<<gGemm, blk, 0, stream>>>(D, Wl2, Wr2, A, B);
    edge_logits   <<<gEdge, blk, 0, stream>>>(A, B, ei, att2, P, MENC);
    edge_exp      <<<gEH,   blk, 0, stream>>>(P, MENC, ei, SSUM);
    edge_aggregate<<<gEdge, blk, 0, stream>>>(A, P, SSUM, ei, C);
    bias_elu      <<<gNF,   blk, 0, stream>>>(C, b2, Y);

    // ---- label prop 2 ----
    hipMemsetAsync(C, 0, NF * sizeof(float), stream);
    lp_scatter<<<gEdgeE, blk, 0, stream>>>(Y, DINV, ei, C);
    lp_update <<<gNF,    blk, 0, stream>>>(C, Y, Ebuf);
    hipMemsetAsync(C, 0, NF * sizeof(float), stream);
    lp_scatter<<<gEdgeE, blk, 0, stream>>>(Ebuf, DINV, ei, C);
    lp_update <<<gNF,    blk, 0, stream>>>(C, Y, Ebuf);

    // ---- mean pool + MLP ----
    pool_count<<<gN,    blk, 0, stream>>>(batch, CNT);
    pool_accum<<<gPool, blk, 0, stream>>>(x, D, Ebuf, batch, POOLS);
    finalize_g<<<(N_GRAPHS * 384 + 255) / 256, blk, 0, stream>>>(POOLS, CNT);
    mlp1<<<(N_GRAPHS * 256 + 255) / 256, blk, 0, stream>>>(POOLS, mW1, mb1, HID);
    mlp2<<<(N_GRAPHS * 128 + 255) / 256, blk, 0, stream>>>(HID, mW2, mb2, out);
}